// LBPEncoder_19791209300512
// MI455X (gfx1250) — compile-verified
//
#include <hip/hip_runtime.h>
#include <cstdint>

// LBP encoder: out[n,k,y,x] = sigmoid(img[n, clamp(y+dy), clamp(x+dx)] - img[n,y,x])
// offsets k=0..7: (-1,-1),(-1,0),(-1,1),(0,-1),(0,1),(1,-1),(1,0),(1,1)
// Memory-bound: 16 MiB in + 128 MiB out -> ~6 us floor at 23.3 TB/s (store-dominated).

#define TILE        64
#define LDS_W       66          // TILE + 2*halo
#define LDS_STRIDE  68          // pad: row base byte offset stays 16B-aligned (68*4=272)
#define THREADS     256         // 8 wave32 waves

#if defined(__has_builtin)
#  if __has_builtin(__builtin_amdgcn_global_load_async_to_lds_b32)
#    define HAVE_ASYNC_LDS 1
#  endif
#  if __has_builtin(__builtin_amdgcn_s_wait_asynccnt)
#    define HAVE_WAIT_ASYNC 1
#  endif
#endif

typedef __attribute__((address_space(1))) int* gptr_i32;
typedef __attribute__((address_space(3))) int* lptr_i32;
typedef float v4f __attribute__((ext_vector_type(4)));   // true vector type for NT store

__device__ __forceinline__ float fast_sigmoid(float x) {
  // sigmoid(x) = 1 / (1 + 2^(-x*log2(e)))  -> v_exp_f32 + v_rcp_f32
  float e = __builtin_amdgcn_exp2f(x * -1.44269504088896340736f);
  return __builtin_amdgcn_rcpf(1.0f + e);
}

__global__ __launch_bounds__(THREADS) void lbp_sigmoid_kernel(
    const float* __restrict__ img, float* __restrict__ out,
    int H, int W) {
  __shared__ float tile[LDS_W * LDS_STRIDE];

  const int n   = blockIdx.z;
  const int ty0 = blockIdx.y * TILE;
  const int tx0 = blockIdx.x * TILE;
  const int tid = threadIdx.x;
  const int HW  = H * W;
  const float* __restrict__ src = img + n * HW;

  // ---- Stage halo tile (66x66, replicate-clamped) into LDS via async copy ----
  for (int i = tid; i < LDS_W * LDS_W; i += THREADS) {
    const int ly = i / LDS_W;
    const int lx = i - ly * LDS_W;
    const int gy = min(max(ty0 + ly - 1, 0), H - 1);
    const int gx = min(max(tx0 + lx - 1, 0), W - 1);
#if defined(HAVE_ASYNC_LDS)
    __builtin_amdgcn_global_load_async_to_lds_b32(
        (gptr_i32)(src + gy * W + gx),
        (lptr_i32)&tile[ly * LDS_STRIDE + lx],
        /*offset=*/0, /*cpol=*/0);
#else
    tile[ly * LDS_STRIDE + lx] = src[gy * W + gx];
#endif
  }
#if defined(HAVE_ASYNC_LDS)
#  if defined(HAVE_WAIT_ASYNC)
  __builtin_amdgcn_s_wait_asynccnt(0);
#  else
  asm volatile("s_wait_asynccnt 0" ::: "memory");
#  endif
#endif
  __syncthreads();

  // ---- Compute: each thread owns 4 consecutive x pixels (16B-aligned stores) ----
  // 16 thread-groups cover one 64-wide row; 256 threads cover 16 rows per pass.
  const int tx4 = tid & 15;            // x group: pixels [4*tx4, 4*tx4+3]
  const int ty  = tid >> 4;            // 0..15
  const int x0  = 4 * tx4;
  const int out_n = n * 8 * HW;

  #pragma unroll
  for (int p = 0; p < 4; ++p) {
    const int y = ty + p * 16;
    // 3 rows x 6 cols neighborhood; row window starts at halo col 4*tx4
    // (byte offset ((y+r)*68 + 4*tx4)*4 is 16B-aligned -> ds_load_b128 + b64)
    float r0[6], r1[6], r2[6];
    const int b0 = (y    ) * LDS_STRIDE + x0;
    const int b1 = (y + 1) * LDS_STRIDE + x0;
    const int b2 = (y + 2) * LDS_STRIDE + x0;
    #pragma unroll
    for (int j = 0; j < 6; ++j) {
      r0[j] = tile[b0 + j];
      r1[j] = tile[b1 + j];
      r2[j] = tile[b2 + j];
    }

    const int obase = out_n + (ty0 + y) * W + (tx0 + x0);

    #pragma unroll
    for (int k = 0; k < 8; ++k) {
      const float* rows[3] = { r0, r1, r2 };
      // offset table: (dy,dx) for k
      const int dy = (k < 3) ? 0 : (k < 5) ? 1 : 2;          // row index into rows[]
      const int dx = (k < 3) ? k : (k == 3) ? 0 : (k == 4) ? 2 : (k - 5);
      const float* nb = rows[dy];
      v4f o;
      o.x = fast_sigmoid(nb[0 + dx] - r1[1]);
      o.y = fast_sigmoid(nb[1 + dx] - r1[2]);
      o.z = fast_sigmoid(nb[2 + dx] - r1[3]);
      o.w = fast_sigmoid(nb[3 + dx] - r1[4]);
      // streamed output, never re-read: non-temporal keeps L2 for the input
      __builtin_nontemporal_store(o, (v4f*)(out + obase + k * HW));
    }
  }
}

extern "C" void kernel_launch(void* const* d_in, const int* in_sizes, int n_in,
                              void* d_out, int out_size, void* d_ws, size_t ws_size,
                              hipStream_t stream) {
  const float* img = (const float*)d_in[0];
  float* out = (float*)d_out;
  const int H = 512, W = 512;
  const int N = in_sizes[0] / (H * W);   // 16
  dim3 grid(W / TILE, H / TILE, N);      // (8, 8, 16)
  lbp_sigmoid_kernel<<<grid, dim3(THREADS), 0, stream>>>(img, out, H, W);
}